// SingleNode_Reserve_SAA_DA_OptNet_46926812676866
// MI455X (gfx1250) — compile-verified
//
#include <hip/hip_runtime.h>
#include <math.h>
#include <stdint.h>

// ---------------- problem constants ----------------
#define G        32
#define T        24
#define NSC      128
#define BBATCH   16
#define WGPB     8            // workgroups per batch element
#define SLOC     16           // scenarios per workgroup (8 * 16 = 128)
#define NTHREADS 256          // 8 waves (wave32)
#define NITER    200          // K_OUT * K_IN
#define GT       768          // G*T
#define VOLLC    1000.0f
#define VOSPC    50.0f

typedef __attribute__((ext_vector_type(2))) float v2f;
typedef __attribute__((ext_vector_type(8))) float v8f;

// ---------------- workspace layout (float units) ----------------
// slots: [parity 2][batch 16][wg 8][2*GT]  (mu/md partial column sums)
#define SLOTS_F   (2 * BBATCH * WGPB * 2 * GT)   // 393216 floats
#define CNT_OFF   (SLOTS_F)                      // 16 x u32 barrier arrival counters
#define GEN_OFF   (SLOTS_F + 16)                 // 16 x u32 barrier generations
#define OBJ_OFF   (SLOTS_F + 32)                 // 128 x f32 per-WG objective partials

// Lock-free inter-workgroup barrier among the 8 WGs of one batch element.
// Works for arbitrary initial 'gen'; 'cnt' is zeroed by the init kernel.
__device__ inline void batch_barrier(unsigned* cnt, unsigned* gen) {
  __threadfence();       // every wave flushes its global stores to L2
  __syncthreads();
  if (threadIdx.x == 0) {
    volatile unsigned* vg = gen;
    unsigned g0 = *vg;
    unsigned arrived = atomicAdd(cnt, 1u);
    if (arrived == (unsigned)(WGPB - 1)) {
      atomicExch(cnt, 0u);
      __threadfence();
      atomicAdd(gen, 1u);
    } else {
      while (*vg == g0) { __builtin_amdgcn_s_sleep(2); }
    }
  }
  __syncthreads();
  __threadfence();
}

__global__ void ws_init_kernel(unsigned* p) {
  if (threadIdx.x < 32) p[threadIdx.x] = 0u;   // cnt[16] + gen[16]
}

__global__ __launch_bounds__(NTHREADS) void pdhg_kernel(
    const float* __restrict__ forecast, const float* __restrict__ omega,
    const float* __restrict__ pminG, const float* __restrict__ pmaxG,
    const float* __restrict__ bcost, const float* __restrict__ ccost,
    float* __restrict__ out, float* __restrict__ ws)
{
  // ---- LDS-resident solver state (~260 KB: CDNA5 320KB/WGP) ----
  __shared__ float duL[SLOC*GT], ddL[SLOC*GT], muL[SLOC*GT], mdL[SLOC*GT];
  __shared__ float y2L[SLOC*T], LSL[SLOC*T], SPL[SLOC*T], omgL[SLOC*T];
  __shared__ float PL[GT], PbL[GT], RuL[GT], RubL[GT], RdL[GT], RdbL[GT];
  __shared__ float l1L[GT], l2L[GT], l3L[GT], l4L[GT];
  __shared__ float musumL[GT], mdsumL[GT];
  __shared__ float y1L[T], fcS[T];
  __shared__ float bS[G], cS[G], pminS[G], pmaxS[G], ruS[G], rdS[G], upNS[G], dnNS[G];
  __shared__ float gscrL[8*GT];     // per-wave G-reduction scratch
  __shared__ float redL[NTHREADS];

  const int tid  = threadIdx.x;
  const int lane = tid & 31;
  const int w    = tid >> 5;
  const int bb   = blockIdx.x / WGPB;
  const int j    = blockIdx.x % WGPB;

  const float tau   = 0.9f / sqrtf(136.0f);   // 0.9/sqrt(N+8)
  const float sigma = tau;
  const float invN  = 1.0f / (float)NSC;

  unsigned* cnt = (unsigned*)(ws + CNT_OFF) + bb;
  unsigned* gen = (unsigned*)(ws + GEN_OFF) + bb;
  float* objpart = ws + OBJ_OFF;

  // ---- constants into LDS ----
  if (tid < G) {
    float bv = bcost[tid];
    bS[tid]    = bv;
    cS[tid]    = ccost[tid];
    pminS[tid] = pminG[tid];
    pmaxS[tid] = pmaxG[tid];
    ruS[tid]   = 0.05f * bv;          // RES_UP * b
    rdS[tid]   = 0.02f * bv;          // RES_DN * b
    upNS[tid]  = 2.0f * bv * invN;    // RT_UP * b / N
    dnNS[tid]  = 0.5f * bv * invN;    // RT_DN * b / N
  }
  if (tid < T) { fcS[tid] = forecast[bb*T + tid]; y1L[tid] = 0.0f; }
  __syncthreads();

  // ---- state init ----
  for (int c0 = tid; c0 < GT; c0 += NTHREADS) {
    int g = c0 / T, t = c0 - g*T;
    float p = fcS[t] * (1.0f / (float)G);
    p = fminf(fmaxf(p, pminS[g]), pmaxS[g]);
    PL[c0] = p;
    RuL[c0] = 0.f; RdL[c0] = 0.f;
    l1L[c0] = 0.f; l2L[c0] = 0.f; l3L[c0] = 0.f; l4L[c0] = 0.f;
  }
  for (int i = tid; i < SLOC*GT; i += NTHREADS) {
    duL[i] = 0.f; ddL[i] = 0.f; muL[i] = 0.f; mdL[i] = 0.f;
  }
  for (int i = tid; i < SLOC*T; i += NTHREADS) {
    y2L[i] = 0.f; LSL[i] = 0.f; SPL[i] = 0.f;
  }
  // omega tile -> LDS via CDNA5 async load-to-LDS (ASYNCcnt path)
  {
    const float* src = omega + ((size_t)bb * NSC + (size_t)j * SLOC) * T;
    for (int i = tid; i < SLOC*T; i += NTHREADS) {
      unsigned ldsoff = (unsigned)(uintptr_t)(&omgL[i]);
      const float* gp = src + i;
      asm volatile("global_load_async_to_lds_b32 %0, %1, off"
                   :: "v"(ldsoff), "v"(gp) : "memory");
    }
    asm volatile("s_wait_asynccnt 0x0" ::: "memory");
  }
  __syncthreads();

  const v2f aones = {1.0f, 1.0f};

  for (int it = 0; it < NITER; ++it) {
    const int parity = it & 1;
    float* slot = ws + ((size_t)(parity * BBATCH + bb) * WGPB + j) * (size_t)(2*GT);

    // ---- Phase A: WMMA column-sums of mu/md over 16 local scenarios ----
    // ones(16x4) x B(4x16) chained over 4 row-chunks => colsum over 16 rows.
    for (int arr = 0; arr < 2; ++arr) {
      const float* M = (arr == 0) ? muL : mdL;
      float* dst = slot + arr * GT;
      for (int i = 0; i < 6; ++i) {            // 48 tiles / 8 waves
        int c0  = (w * 6 + i) * 16;
        int col = c0 + (lane & 15);
        v8f acc = {0.f,0.f,0.f,0.f,0.f,0.f,0.f,0.f};
        #pragma unroll
        for (int kb = 0; kb < 4; ++kb) {
          int rx = kb * 4 + ((lane & 16) ? 2 : 0);
          v2f bv;
          bv.x = M[rx * GT + col];
          bv.y = M[(rx + 1) * GT + col];
          acc = __builtin_amdgcn_wmma_f32_16x16x4_f32(
              false, aones, false, bv, (short)0, acc, false, false);
        }
        if (lane < 16) dst[c0 + lane] = acc[0];   // D row M=0 = colsums
      }
    }

    // ---- Phase B: inter-WG barrier (slots of this parity are complete) ----
    batch_barrier(cnt, gen);

    // ---- Phase C: reduce the 8 WG partials -> full mu/md sums (replicated) ----
    {
      const float* base = ws + (size_t)(parity * BBATCH + bb) * WGPB * (size_t)(2*GT);
      for (int cc = tid; cc < 2*GT; cc += NTHREADS) {
        float s = 0.f;
        #pragma unroll
        for (int jj = 0; jj < WGPB; ++jj) s += base[jj * (2*GT) + cc];
        if (cc < GT) musumL[cc] = s; else mdsumL[cc - GT] = s;
      }
    }
    __syncthreads();

    // ---- Phase D: shared primal + l1/l2 dual updates (replicated) ----
    for (int c0 = tid; c0 < GT; c0 += NTHREADS) {
      int g = c0 / T, t = c0 - g*T;
      float Po = PL[c0], l1o = l1L[c0], l2o = l2L[c0];
      float dl3 = (t > 0 ? l3L[c0-1] : 0.f) - (t < T-1 ? l3L[c0] : 0.f);
      float dl4 = (t > 0 ? l4L[c0-1] : 0.f) - (t < T-1 ? l4L[c0] : 0.f);
      float Pn = Po - tau * (bS[g] + y1L[t] + l1o - l2o + dl3 - dl4);
      Pn = fminf(fmaxf(Pn, pminS[g]), pmaxS[g]);
      float Pb = 2.f * Pn - Po;
      PL[c0] = Pn; PbL[c0] = Pb;
      float Ruo = RuL[c0];
      float Run = fmaxf(Ruo - tau * (ruS[g] + l1o - musumL[c0]), 0.f);
      float Rub = 2.f * Run - Ruo;
      RuL[c0] = Run; RubL[c0] = Rub;
      float Rdo = RdL[c0];
      float Rdn = fmaxf(Rdo - tau * (rdS[g] + l2o - mdsumL[c0]), 0.f);
      float Rdb = 2.f * Rdn - Rdo;
      RdL[c0] = Rdn; RdbL[c0] = Rdb;
      l1L[c0] = fmaxf(l1o + sigma * (Pb + Rub - pmaxS[g]), 0.f);
      l2L[c0] = fmaxf(l2o + sigma * (pminS[g] - Pb + Rdb), 0.f);
    }
    __syncthreads();

    // ---- Phase E: ramp duals + y1 (replicated) ----
    for (int c0 = tid; c0 < GT; c0 += NTHREADS) {
      int g = c0 / T, t = c0 - g*T;
      if (t < T-1) {
        float diff = PbL[c0+1] - PbL[c0];
        l3L[c0] = fmaxf(l3L[c0] + sigma * ( diff - pmaxS[g]), 0.f);  // ramp = pmax
        l4L[c0] = fmaxf(l4L[c0] + sigma * (-diff - pmaxS[g]), 0.f);
      }
    }
    if (tid < T) {
      float s = 0.f;
      #pragma unroll
      for (int g = 0; g < G; ++g) s += PbL[g*T + tid];
      y1L[tid] += sigma * (s - fcS[tid]);
    }

    // ---- Phase F: per-scenario updates (wave w owns scenarios w and w+8, lane = g) ----
    {
      float upg = upNS[lane], dng = dnNS[lane];
      for (int si = 0; si < 2; ++si) {
        int sn  = w + 8 * si;
        int row = sn * GT + lane * T;
        int cgt = lane * T;
        float gdiff[T];
        #pragma unroll
        for (int t = 0; t < T; ++t) {
          float y2o = y2L[sn*T + t];
          float muo = muL[row+t], mdo = mdL[row+t];
          float duo = duL[row+t], ddo = ddL[row+t];
          float dun = fmaxf(duo - tau * (upg + y2o + muo), 0.f);
          float ddn = fmaxf(ddo - tau * (dng - y2o + mdo), 0.f);
          float dub = 2.f*dun - duo, ddb = 2.f*ddn - ddo;
          duL[row+t] = dun; ddL[row+t] = ddn;
          muL[row+t] = fmaxf(muo + sigma * (dub - RubL[cgt+t]), 0.f);
          mdL[row+t] = fmaxf(mdo + sigma * (ddb - RdbL[cgt+t]), 0.f);
          gdiff[t] = dub - ddb;
        }
        __syncthreads();
        #pragma unroll
        for (int t = 0; t < T; ++t) gscrL[w*GT + lane*T + t] = gdiff[t];
        __syncthreads();
        if (lane < T) {
          int t = lane;
          float gs = 0.f;
          #pragma unroll
          for (int g = 0; g < G; ++g) gs += gscrL[w*GT + g*T + t];
          float y2o = y2L[sn*T + t];
          float LSo = LSL[sn*T + t], SPo = SPL[sn*T + t];
          float LSn = fmaxf(LSo - tau * (VOLLC*invN + y2o), 0.f);
          float SPn = fmaxf(SPo - tau * (VOSPC*invN - y2o), 0.f);
          float LSb = 2.f*LSn - LSo, SPb = 2.f*SPn - SPo;
          LSL[sn*T + t] = LSn; SPL[sn*T + t] = SPn;
          y2L[sn*T + t] = y2o + sigma * (gs + LSb - SPb - omgL[sn*T + t]);
        }
        __syncthreads();
      }
    }
  }

  // ---- objective ----
  float part = 0.f;
  for (int i = tid; i < SLOC*GT; i += NTHREADS) {
    int cell = i % GT;
    int g = cell / T;
    part += upNS[g] * duL[i] + dnNS[g] * ddL[i];   // already divided by N
  }
  for (int i = tid; i < SLOC*T; i += NTHREADS)
    part += (VOLLC*invN) * LSL[i] + (VOSPC*invN) * SPL[i];
  if (j == 0) {
    for (int c0 = tid; c0 < GT; c0 += NTHREADS) {
      int g = c0 / T;
      part += bS[g]*PL[c0] + ruS[g]*RuL[c0] + rdS[g]*RdL[c0];
    }
    if (tid == 0) {
      float csum = 0.f;
      for (int g = 0; g < G; ++g) csum += cS[g];
      part += (float)T * csum;
    }
  }
  redL[tid] = part;
  __syncthreads();
  if (tid == 0) {
    float s = 0.f;
    for (int i = 0; i < NTHREADS; ++i) s += redL[i];
    objpart[bb*WGPB + j] = s;
  }
  batch_barrier(cnt, gen);
  if (j == 0 && tid == 0) {
    float o = 0.f;
    for (int jj = 0; jj < WGPB; ++jj) o += objpart[bb*WGPB + jj];
    out[bb] = o;
  }
}

extern "C" void kernel_launch(void* const* d_in, const int* in_sizes, int n_in,
                              void* d_out, int out_size, void* d_ws, size_t ws_size,
                              hipStream_t stream) {
  (void)in_sizes; (void)n_in; (void)out_size; (void)ws_size;
  const float* forecast = (const float*)d_in[0];
  const float* omega    = (const float*)d_in[1];
  const float* pmin     = (const float*)d_in[2];
  const float* pmax     = (const float*)d_in[3];
  const float* b        = (const float*)d_in[4];
  const float* c        = (const float*)d_in[5];
  float* ws  = (float*)d_ws;
  float* out = (float*)d_out;

  ws_init_kernel<<<1, 64, 0, stream>>>((unsigned*)(ws + CNT_OFF));
  pdhg_kernel<<<BBATCH * WGPB, NTHREADS, 0, stream>>>(
      forecast, omega, pmin, pmax, b, c, out, ws);
}